// ViewTransform_8418135900339
// MI455X (gfx1250) — compile-verified
//
#include <hip/hip_runtime.h>

#define BEVN 100
#define FH 88
#define FW 176
#define NPIX (FH * FW)   // 15488 floats = 61952 bytes of LDS per block

#if defined(__has_builtin)
#  if __has_builtin(__builtin_amdgcn_global_load_async_to_lds_b128) && \
      __has_builtin(__builtin_amdgcn_s_wait_asynccnt)
#    define USE_ASYNC_LDS 1
#  endif
#endif
#ifndef USE_ASYNC_LDS
#  define USE_ASYNC_LDS 0
#endif

#if USE_ASYNC_LDS
typedef int v4i __attribute__((vector_size(16)));
typedef __attribute__((address_space(1))) v4i GV4;   // global int4
typedef __attribute__((address_space(3))) v4i LV4;   // LDS int4
#endif

__global__ __launch_bounds__(256)
void ViewTransform_bev_kernel(const float* __restrict__ img,
                              const float* __restrict__ intr,
                              const int* __restrict__ p_imgw,
                              const int* __restrict__ p_imgh,
                              float* __restrict__ out,
                              int C)
{
    __shared__ float smem[NPIX];

    const int c = blockIdx.x;
    const int b = blockIdx.y;
    const float* __restrict__ src = img + ((size_t)(b * C + c)) * NPIX;

    // ---- Stage the 88x176 feature slice for this (b,c) into LDS ----
#if USE_ASYNC_LDS
    for (int i = threadIdx.x * 4; i < NPIX; i += 256 * 4) {
        __builtin_amdgcn_global_load_async_to_lds_b128(
            (GV4*)(src + i), (LV4*)&smem[i], 0, 0);
    }
    __builtin_amdgcn_s_wait_asynccnt(0);
#else
    for (int i = threadIdx.x * 4; i < NPIX; i += 256 * 4) {
        float4 v = *(const float4*)(src + i);
        *(float4*)&smem[i] = v;
    }
#endif
    __syncthreads();

    const float fimgw = (float)(*p_imgw);   // 704
    const float fimgh = (float)(*p_imgh);   // 352
    const float fx = intr[b * 9 + 0];
    const float cx = intr[b * 9 + 2];
    const float fy = intr[b * 9 + 4];
    const float cy = intr[b * 9 + 5];

    float* __restrict__ dst = out + ((size_t)(b * C + c)) * (BEVN * BEVN);

    for (int cell = threadIdx.x; cell < BEVN * BEVN; cell += 256) {
        const int zi = cell / BEVN;
        const int xi = cell - zi * BEVN;

        const float xw   = -25.0f + 0.125f + 0.5f * (float)xi;
        const float z    = 0.125f + 0.5f * (float)zi;
        const float invz = 1.0f / z;
        const float xz   = xw * invz;

        // u shared by all 3 planes; unnormalize to feature-map pixel coords
        const float u  = ((fx * xz + cx) / fimgw - 0.5f) * 2.0f;
        const float px = ((u + 1.0f) * (float)FW - 1.0f) * 0.5f;
        const float x0f = floorf(px);
        const int   x0  = (int)x0f;
        const int   x1  = x0 + 1;
        const float wx1 = px - x0f;
        const float wx0 = 1.0f - wx1;
        const bool  vx0 = (x0 >= 0) & (x0 < FW);
        const bool  vx1 = (x1 >= 0) & (x1 < FW);
        const int   cx0 = min(max(x0, 0), FW - 1);
        const int   cx1 = min(max(x1, 0), FW - 1);

        float acc = 0.0f;
        const float plane_y[3] = { -1.0f, 0.5f, 2.0f };
        #pragma unroll
        for (int k = 0; k < 3; ++k) {
            const float yzv = plane_y[k] * invz;
            const float v   = ((fy * yzv + cy) / fimgh - 0.5f) * 2.0f;
            const float py  = ((v + 1.0f) * (float)FH - 1.0f) * 0.5f;
            const float y0f = floorf(py);
            const int   y0  = (int)y0f;
            const int   y1  = y0 + 1;
            const float wy1 = py - y0f;
            const float wy0 = 1.0f - wy1;
            const bool  vy0 = (y0 >= 0) & (y0 < FH);
            const bool  vy1 = (y1 >= 0) & (y1 < FH);
            const int   cy0 = min(max(y0, 0), FH - 1);
            const int   cy1 = min(max(y1, 0), FH - 1);

            const float s00 = (vx0 & vy0) ? smem[cy0 * FW + cx0] : 0.0f;
            const float s10 = (vx1 & vy0) ? smem[cy0 * FW + cx1] : 0.0f;
            const float s01 = (vx0 & vy1) ? smem[cy1 * FW + cx0] : 0.0f;
            const float s11 = (vx1 & vy1) ? smem[cy1 * FW + cx1] : 0.0f;

            acc += (wx0 * wy0) * s00 + (wx1 * wy0) * s10
                 + (wx0 * wy1) * s01 + (wx1 * wy1) * s11;
        }
        dst[cell] = acc * (1.0f / 3.0f);
    }
}

extern "C" void kernel_launch(void* const* d_in, const int* in_sizes, int n_in,
                              void* d_out, int out_size, void* d_ws, size_t ws_size,
                              hipStream_t stream) {
    const float* img  = (const float*)d_in[0];
    const float* intr = (const float*)d_in[1];
    const int*   imgw = (const int*)d_in[2];
    const int*   imgh = (const int*)d_in[3];
    float*       out  = (float*)d_out;

    const int B = in_sizes[1] / 9;                                  // 8
    const int C = (int)(in_sizes[0] / ((long long)B * NPIX));       // 512

    dim3 grid(C, B, 1);
    dim3 block(256, 1, 1);
    hipLaunchKernelGGL(ViewTransform_bev_kernel, grid, block, 0, stream,
                       img, intr, imgw, imgh, out, C);
}